// TwoLayerSNN_52003464020392
// MI455X (gfx1250) — compile-verified
//
#include <hip/hip_runtime.h>

// CDNA5 / gfx1250, wave32. Output-bandwidth-bound SNN unroll:
//   ~335 MB of streaming stores -> ~14.6 us roofline at 23.3 TB/s.
// fc2 (binary spikes [32x10] @ w2^T [10x2]) is done per-wave with
// V_WMMA_F32_16X16X4_F32 (full f32 precision), K=10 padded to 12,
// chunked 4+4+4, accumulator chained through C.

typedef __attribute__((ext_vector_type(2))) float v2f;
typedef __attribute__((ext_vector_type(8))) float v8f;

#define SNN_BETA 0.8f
#define SNN_NSTEPS 20

__global__ __launch_bounds__(256) void TwoLayerSNN_52003464020392_kernel(
    const float* __restrict__ xp,
    const float* __restrict__ w1,   // [10]
    const float* __restrict__ b1,   // [10]
    const float* __restrict__ w2,   // [2][10] row-major
    const float* __restrict__ b2,   // [2]
    float* __restrict__ outp,       // spk2_rec [20,B,2] then mem2_rec [20,B,2]
    int n)
{
    // Per-wave staging: spike tile (stride 12 -> bank-conflict-free) + cur2.
    __shared__ float lds_spk[8][32][12];   // 12 KB
    __shared__ float lds_cur2[8][32][2];   //  2 KB

    const int tid   = threadIdx.x;
    const int lane  = tid & 31;
    const int wave  = tid >> 5;
    const int row   = lane & 15;    // M-row / N-col within 16-lane half
    const int khalf = lane >> 4;    // selects K rows {0,1} vs {2,3} in A/B frags
    const long long gid = (long long)blockIdx.x * 256 + tid;
    const long long B   = (long long)n;

    const float xv = (gid < B) ? xp[gid] : 0.0f;

    // Layer-1 constants + state (registers).
    float cur1[10], mem1[10];
#pragma unroll
    for (int j = 0; j < 10; ++j) {
        cur1[j] = fmaf(xv, w1[j], b1[j]);   // fc1: x*w1 + b1 (time-invariant)
        mem1[j] = 0.0f;
    }
    float mem2c0 = 0.0f, mem2c1 = 0.0f;
    const float b2_0 = b2[0], b2_1 = b2[1];

    // B-matrix fragments for V_WMMA_F32_16X16X4_F32 (time-invariant).
    // 4x16 f32 B, 2 VGPRs: VGPR0 holds row K=2*khalf, VGPR1 row K=2*khalf+1,
    // lane's column N = lane&15. Only columns N=0,1 of D are consumed, so only
    // those B columns carry w2; everything else (incl. K>=10 padding) is zero.
    v2f bfrag[3];
    {
        const int c = row;
#pragma unroll
        for (int q = 0; q < 3; ++q) {
            const int kx = 4 * q + 2 * khalf;
            const int ky = kx + 1;
            float bx = 0.0f, by = 0.0f;
            if (c < 2) {
                if (kx < 10) bx = w2[c * 10 + kx];
                if (ky < 10) by = w2[c * 10 + ky];
            }
            bfrag[q].x = bx;
            bfrag[q].y = by;
        }
    }

    // Zero the K-padding slots (j = 10, 11) once; steps only rewrite j < 10.
    lds_spk[wave][lane][10] = 0.0f;
    lds_spk[wave][lane][11] = 0.0f;

    v2f* out_spk = (v2f*)outp;                         // [20*B] float2
    v2f* out_mem = out_spk + (long long)SNN_NSTEPS * B;

#pragma unroll 1
    for (int t = 0; t < SNN_NSTEPS; ++t) {
        // ---- layer 1 LIF (reset_mechanism='subtract', reset from prev mem) ----
#pragma unroll
        for (int j = 0; j < 10; ++j) {
            const float r = (mem1[j] - 1.0f > 0.0f) ? 1.0f : 0.0f;
            mem1[j] = fmaf(SNN_BETA, mem1[j], cur1[j]) - r;
            lds_spk[wave][lane][j] = (mem1[j] - 1.0f > 0.0f) ? 1.0f : 0.0f;
        }

        // ---- fc2 via WMMA: cur2[32x2] = spk1[32x10] @ w2^T[10x2] ----
        // Two 16-sample tiles per wave; 3 K-chunks chained through C.
        // Intra-wave LDS ops are in-order -> no barrier needed.
#pragma unroll
        for (int h = 0; h < 2; ++h) {
            v8f d = {};
#pragma unroll
            for (int q = 0; q < 3; ++q) {
                // A 16x4 f32 layout (ISA 7.12.2): VGPR0 K=2*khalf, VGPR1 K=2*khalf+1.
                v2f a;
                a.x = lds_spk[wave][h * 16 + row][4 * q + 2 * khalf];
                a.y = lds_spk[wave][h * 16 + row][4 * q + 2 * khalf + 1];
                d = __builtin_amdgcn_wmma_f32_16x16x4_f32(
                        /*neg_a=*/false, a, /*neg_b=*/false, bfrag[q],
                        /*c_mod=*/(short)0, d, /*reuse_a=*/false, /*reuse_b=*/false);
            }
            // D layout: VGPR v -> M = v + 8*khalf, N = lane&15. Scatter the two
            // useful columns (N=0,1) back so every lane can read its own cur2.
            if (row < 2) {
#pragma unroll
                for (int v = 0; v < 8; ++v)
                    lds_cur2[wave][h * 16 + 8 * khalf + v][row] = d[v];
            }
        }
        const float cur2_0 = lds_cur2[wave][lane][0] + b2_0;
        const float cur2_1 = lds_cur2[wave][lane][1] + b2_1;

        // ---- layer 2 LIF ----
        const float r0 = (mem2c0 - 1.0f > 0.0f) ? 1.0f : 0.0f;
        mem2c0 = fmaf(SNN_BETA, mem2c0, cur2_0) - r0;
        const float r1 = (mem2c1 - 1.0f > 0.0f) ? 1.0f : 0.0f;
        mem2c1 = fmaf(SNN_BETA, mem2c1, cur2_1) - r1;
        const float s0 = (mem2c0 - 1.0f > 0.0f) ? 1.0f : 0.0f;
        const float s1 = (mem2c1 - 1.0f > 0.0f) ? 1.0f : 0.0f;

        // ---- record spk2/mem2: coalesced 8B non-temporal streaming stores ----
        if (gid < B) {
            v2f sv; sv.x = s0;     sv.y = s1;
            v2f mv; mv.x = mem2c0; mv.y = mem2c1;
            const long long o = (long long)t * B + gid;
            __builtin_nontemporal_store(sv, &out_spk[o]);
            __builtin_nontemporal_store(mv, &out_mem[o]);
        }
    }
}

extern "C" void kernel_launch(void* const* d_in, const int* in_sizes, int n_in,
                              void* d_out, int out_size, void* d_ws, size_t ws_size,
                              hipStream_t stream) {
    (void)n_in; (void)out_size; (void)d_ws; (void)ws_size;
    const float* x  = (const float*)d_in[0];  // [B,1]
    const float* w1 = (const float*)d_in[1];  // [10,1]
    const float* b1 = (const float*)d_in[2];  // [10]
    const float* w2 = (const float*)d_in[3];  // [2,10]
    const float* b2 = (const float*)d_in[4];  // [2]
    const int n = in_sizes[0];                // B = 1,048,576 (multiple of 256)
    const int blocks = (n + 255) / 256;
    TwoLayerSNN_52003464020392_kernel<<<blocks, 256, 0, stream>>>(
        x, w1, b1, w2, b2, (float*)d_out, n);
}